// MyModel_61933428415922
// MI455X (gfx1250) — compile-verified
//
#include <hip/hip_runtime.h>
#include <hip/hip_bf16.h>

// ---------------------------------------------------------------------------
// Windowed local-attention scores on gfx1250 (MI455X), fp32 WMMA path.
//
// Stage 1: fused 1x1-conv projection GEMM. For each batch b:
//   W_eff (256x256) x X (256x36864) -> QK plane (256x36864) in workspace.
//   Rows 0..127   = q_w
//   Rows 128..255 = kv_w rows h*48+d (d<16)   (the V part of kv is dead code
//                                              in the reference output)
// Stage 2: per (batch-head, 8x8 block): S = Q(64x16) . K(144x16)^T -> 64x144,
//   K gathered from the 12x12 halo window with zero padding at borders.
//
// Both matmuls use V_WMMA_F32_16X16X4_F32 (wave32, 16x16 fp32 tiles).
// Memory-bound: 679 MB output write (~29 us at 23.3 TB/s) >> 25 GFLOP compute.
// Workspace: 4*256*36864 floats = 151 MB.
// ---------------------------------------------------------------------------

typedef __attribute__((ext_vector_type(2))) float v2f;
typedef __attribute__((ext_vector_type(8))) float v8f;

#define HW   (192*192)   // 36864 pixels per plane
#define NB   576         // 24*24 blocks
#define NHW  24

// ---------------------------------------------------------------------------
// Stage 1: projection GEMM.  grid = B(4) * Mtiles(16) * PixGroups(288),
// block = 128 threads (4 waves).  Wave computes a 16(M) x 32(N) tile:
// 64 K-steps x 2 wmma.
// ---------------------------------------------------------------------------
__global__ __launch_bounds__(128) void proj_kernel(
    const float* __restrict__ x,     // (4,256,192,192)
    const float* __restrict__ q_w,   // (128,256)
    const float* __restrict__ kv_w,  // (384,256)
    float* __restrict__ qk)          // ws: (4,256,36864)
{
    const int wg = blockIdx.x;
    const int ng = wg % 288;                 // group of 128 pixels
    const int mt = (wg / 288) & 15;          // M tile (16 output channels)
    const int b  = wg / (288 * 16);

    const int lane = threadIdx.x & 31;
    const int wave = threadIdx.x >> 5;
    const int g    = lane >> 4;              // K half-select (0/1)
    const int m    = lane & 15;              // row/col within tile

    // This lane's weight row (A matrix row M = mt*16 + m).
    const int mo = mt * 16 + m;
    const float* wrow;
    if (mo < 128) {
        wrow = q_w + (size_t)mo * 256;
    } else {
        const int h = (mo - 128) >> 4;
        const int d = (mo - 128) & 15;
        wrow = kv_w + (size_t)(h * 48 + d) * 256;   // skip dead V rows
    }

    const int pixbase = ng * 128 + wave * 32;       // 32 pixels per wave
    const float* xb = x + (size_t)b * 256 * HW;

    v8f c0 = {};
    v8f c1 = {};
    const int p0 = pixbase + m;
    const int p1 = p0 + 16;

    for (int k0 = 0; k0 < 256; k0 += 4) {
        const int ka = k0 + 2 * g;                  // even -> 8B aligned
        const v2f a = *(const v2f*)(wrow + ka);

        const float* xk0 = xb + (size_t)ka * HW;
        const float* xk1 = xk0 + HW;
        v2f b0; b0.x = xk0[p0]; b0.y = xk1[p0];
        v2f b1; b1.x = xk0[p1]; b1.y = xk1[p1];

        c0 = __builtin_amdgcn_wmma_f32_16x16x4_f32(
                 false, a, false, b0, (short)0, c0, false, false);
        c1 = __builtin_amdgcn_wmma_f32_16x16x4_f32(
                 false, a, false, b1, (short)0, c1, false, false);
    }

    // D layout: reg r, lane -> M = r + 8*g, N = m.
    float* outp = qk + ((size_t)b * 256 + mt * 16) * HW + pixbase;
    #pragma unroll
    for (int r = 0; r < 8; ++r) {
        const int mm = r + 8 * g;
        outp[(size_t)mm * HW + m]      = c0[r];
        outp[(size_t)mm * HW + m + 16] = c1[r];
    }
}

// ---------------------------------------------------------------------------
// Stage 2: attention scores.  grid = 32*576 blocks (one per (bh, n)),
// block = 128 threads (4 waves).  Wave w handles queries w*16..w*16+15,
// loops over 9 key tiles of 16; 4 wmma (K=16) per tile.
// ---------------------------------------------------------------------------
__global__ __launch_bounds__(128) void attn_kernel(
    const float* __restrict__ ws,   // (4,256,36864): [0..127]=Q, [128..255]=K
    float* __restrict__ out)        // (32,576,64,144)
{
    const int blk = blockIdx.x;          // bh*576 + n
    const int n   = blk % NB;
    const int bh  = blk / NB;
    const int b   = bh >> 3;
    const int h   = bh & 7;
    const int bi  = n / NHW;
    const int bj  = n % NHW;

    const int lane = threadIdx.x & 31;
    const int w    = threadIdx.x >> 5;
    const int g    = lane >> 4;
    const int m    = lane & 15;

    const float* qbase = ws + ((size_t)b * 256 + h * 16) * HW;
    const float* kbase = ws + ((size_t)b * 256 + 128 + h * 16) * HW;

    // ---- Q operand (A matrix): query qi = w*16 + m, 16 dims in 4 K-steps.
    const int qi  = w * 16 + m;
    const int pix = (bi * 8 + (qi >> 3)) * 192 + bj * 8 + (qi & 7);
    v2f a[4];
    #pragma unroll
    for (int s = 0; s < 4; ++s) {
        const int ka = 4 * s + 2 * g;
        a[s].x = qbase[(size_t)ka * HW + pix];
        a[s].y = qbase[(size_t)(ka + 1) * HW + pix];
    }

    float* oblk = out + (size_t)blk * 64 * 144 + (size_t)w * 16 * 144;

    #pragma unroll
    for (int t = 0; t < 9; ++t) {
        // ---- K operand (B matrix): key kj = t*16 + m in the 12x12 window.
        const int kj = t * 16 + m;                 // < 144
        const int wi = kj / 12;
        const int wj = kj - wi * 12;
        const int hh = bi * 8 + wi - 2;            // unpadded coords
        const int ww = bj * 8 + wj - 2;
        const bool valid = ((unsigned)hh < 192u) && ((unsigned)ww < 192u);
        // Clamp address so every lane issues an in-bounds load (EXEC stays
        // all-1s); zero-fill via cndmask for halo lanes.
        const size_t kp = valid ? ((size_t)hh * 192 + ww) : 0;

        v8f c = {};
        #pragma unroll
        for (int s = 0; s < 4; ++s) {
            const int kb = 4 * s + 2 * g;
            const float f0 = kbase[(size_t)kb * HW + kp];
            const float f1 = kbase[(size_t)(kb + 1) * HW + kp];
            v2f bv;
            bv.x = valid ? f0 : 0.0f;
            bv.y = valid ? f1 : 0.0f;
            c = __builtin_amdgcn_wmma_f32_16x16x4_f32(
                    false, a[s], false, bv, (short)0, c, false, false);
        }

        // D layout: reg r -> query row (r + 8*g) within tile, key col kj.
        #pragma unroll
        for (int r = 0; r < 8; ++r) {
            oblk[(size_t)(r + 8 * g) * 144 + kj] = c[r];
        }
    }
}

// ---------------------------------------------------------------------------
extern "C" void kernel_launch(void* const* d_in, const int* in_sizes, int n_in,
                              void* d_out, int out_size, void* d_ws, size_t ws_size,
                              hipStream_t stream)
{
    const float* x    = (const float*)d_in[0];   // (4,256,192,192)
    const float* q_w  = (const float*)d_in[1];   // (128,256)
    const float* kv_w = (const float*)d_in[2];   // (384,256)
    float* out = (float*)d_out;                  // (32,576,64,144)
    float* ws  = (float*)d_ws;                   // >= 151 MB

    // Stage 1: 4 batches * 16 M-tiles * 288 pixel-groups = 18432 WGs.
    proj_kernel<<<4 * 16 * 288, 128, 0, stream>>>(x, q_w, kv_w, ws);
    // Stage 2: 32 batch-heads * 576 blocks = 18432 WGs.
    attn_kernel<<<32 * 576, 128, 0, stream>>>(ws, out);
}